// SGLDMExtractor_30571577213262
// MI455X (gfx1250) — compile-verified
//
#include <hip/hip_runtime.h>
#include <hip/hip_bf16.h>
#include <math.h>

#define B_IMG   128
#define H       512
#define W       512
#define ROWS    16
#define SLICES  (H / ROWS)       // 32
#define THREADS 256
#define N_PIX   (H * W)          // 262144

// Low 32 bits of a generic pointer to a __shared__ object are the wave-relative
// LDS byte offset (flat->LDS mapping uses addr[31:0]); that's exactly what the
// async-to-LDS VDST operand wants.
__device__ __forceinline__ unsigned lds_off(const void* p) {
  return (unsigned)(unsigned long long)p;
}

// Issue async global->LDS b128 transfers for `bytes` bytes (16B multiple).
// Tracked by ASYNCcnt; caller must async_wait_barrier() before reading LDS.
__device__ __forceinline__ void async_copy_issue(const float* __restrict__ gbase,
                                                 float* tile, unsigned bytes) {
  const unsigned base = lds_off(tile);
  for (unsigned ofs = threadIdx.x * 16u; ofs < bytes; ofs += THREADS * 16u) {
    unsigned laddr = base + ofs;
    const void* g = (const char*)gbase + ofs;
    asm volatile("global_load_async_to_lds_b128 %0, %1, off"
                 :: "v"(laddr), "v"(g) : "memory");
  }
}

__device__ __forceinline__ void async_wait_barrier() {
#if __has_builtin(__builtin_amdgcn_s_wait_asynccnt)
  __builtin_amdgcn_s_wait_asynccnt(0);
#else
  asm volatile("s_wait_asynccnt 0" ::: "memory");
#endif
  __syncthreads();
}

// Pass 1: per-(image,slice) min/max of d = |img - shifted|.
__global__ void k_minmax(const float* __restrict__ x, float* __restrict__ mm) {
  __shared__ float tile[(ROWS + 1) * W];
  __shared__ float rmin[THREADS];
  __shared__ float rmax[THREADS];

  const int blk = blockIdx.x;
  const int img = blk >> 5;            // / SLICES
  const int sl  = blk & (SLICES - 1);
  const int r0  = sl * ROWS;
  const int nrows = (r0 + ROWS + 1 <= H) ? (ROWS + 1) : ROWS;  // halo row if it exists

  const float* gbase = x + (size_t)img * N_PIX + (size_t)r0 * W;
  async_copy_issue(gbase, tile, (unsigned)nrows * W * 4u);
  async_wait_barrier();

  float vmin = 3.402823466e38f, vmax = 0.0f;  // d >= 0
  #pragma unroll 4
  for (int e = threadIdx.x; e < ROWS * W; e += THREADS) {
    const int c  = e & (W - 1);
    const int gr = r0 + (e >> 9);      // e / W
    const float a = tile[e];
    const float b = (gr < H - 1 && c < W - 1) ? tile[e + W + 1] : 0.0f;
    const float d = fabsf(a - b);
    vmin = fminf(vmin, d);
    vmax = fmaxf(vmax, d);
  }
  rmin[threadIdx.x] = vmin;
  rmax[threadIdx.x] = vmax;
  __syncthreads();
  for (int s = THREADS / 2; s > 0; s >>= 1) {
    if ((int)threadIdx.x < s) {
      rmin[threadIdx.x] = fminf(rmin[threadIdx.x], rmin[threadIdx.x + s]);
      rmax[threadIdx.x] = fmaxf(rmax[threadIdx.x], rmax[threadIdx.x + s]);
    }
    __syncthreads();
  }
  if (threadIdx.x == 0) {
    mm[blk * 2 + 0] = rmin[0];
    mm[blk * 2 + 1] = rmax[0];
  }
}

// Pass 2: normalized sums S1, S2, S3 per (image,slice). Re-read hits L2
// (128 MB tensor < 192 MB L2). Async DMA is issued first so the 64 uniform
// scalar loads of the min/max partials overlap the in-flight copy.
__global__ void k_sums(const float* __restrict__ x, const float* __restrict__ mm,
                       float* __restrict__ sums) {
  __shared__ float tile[(ROWS + 1) * W];
  __shared__ float r1[THREADS];
  __shared__ float r2[THREADS];
  __shared__ float r3[THREADS];

  const int blk = blockIdx.x;
  const int img = blk >> 5;
  const int sl  = blk & (SLICES - 1);
  const int r0  = sl * ROWS;
  const int nrows = (r0 + ROWS + 1 <= H) ? (ROWS + 1) : ROWS;

  const float* gbase = x + (size_t)img * N_PIX + (size_t)r0 * W;
  async_copy_issue(gbase, tile, (unsigned)nrows * W * 4u);

  // Combine the 32 per-slice min/max pairs while the DMA is in flight.
  float dmin = 3.402823466e38f, dmax = 0.0f;
  #pragma unroll
  for (int i = 0; i < SLICES; ++i) {
    dmin = fminf(dmin, mm[(img * SLICES + i) * 2 + 0]);
    dmax = fmaxf(dmax, mm[(img * SLICES + i) * 2 + 1]);
  }
  const float inv = 1.0f / (dmax - dmin + 1e-6f);

  async_wait_barrier();

  float s1 = 0.0f, s2 = 0.0f, s3 = 0.0f;
  #pragma unroll 4
  for (int e = threadIdx.x; e < ROWS * W; e += THREADS) {
    const int c  = e & (W - 1);
    const int gr = r0 + (e >> 9);
    const float a = tile[e];
    const float b = (gr < H - 1 && c < W - 1) ? tile[e + W + 1] : 0.0f;
    const float d  = fabsf(a - b);
    const float dn = (d - dmin) * inv;
    s1 += dn;
    s2 += dn * dn;
    s3 += dn * __logf(dn + 1e-8f);
  }
  r1[threadIdx.x] = s1;
  r2[threadIdx.x] = s2;
  r3[threadIdx.x] = s3;
  __syncthreads();
  for (int s = THREADS / 2; s > 0; s >>= 1) {
    if ((int)threadIdx.x < s) {
      r1[threadIdx.x] += r1[threadIdx.x + s];
      r2[threadIdx.x] += r2[threadIdx.x + s];
      r3[threadIdx.x] += r3[threadIdx.x + s];
    }
    __syncthreads();
  }
  if (threadIdx.x == 0) {
    sums[blk * 3 + 0] = r1[0];
    sums[blk * 3 + 1] = r2[0];
    sums[blk * 3 + 2] = r3[0];
  }
}

// Pass 3: combine slice partials, emit the 6 features per image.
__global__ void k_final(const float* __restrict__ sums, float* __restrict__ out) {
  const int b = threadIdx.x;
  if (b >= B_IMG) return;
  float S1 = 0.0f, S2 = 0.0f, S3 = 0.0f;
  for (int s = 0; s < SLICES; ++s) {
    const float* p = sums + (size_t)(b * SLICES + s) * 3;
    S1 += p[0];
    S2 += p[1];
    S3 += p[2];
  }
  const float N = (float)N_PIX;
  const float mean     = S1 / N;
  const float contrast = fmaxf(S2 - S1 * S1 / N, 0.0f);  // sum((d-mean)^2)
  const float var      = contrast / (N - 1.0f);           // ddof=1
  const float stdd     = sqrtf(var);
  const float energy   = S2;
  const float entropy  = -S3;
  const float smooth   = 1.0f - 1.0f / (1.0f + var);
  out[b * 6 + 0] = mean;
  out[b * 6 + 1] = stdd;
  out[b * 6 + 2] = energy;
  out[b * 6 + 3] = entropy;
  out[b * 6 + 4] = contrast;
  out[b * 6 + 5] = smooth;
}

extern "C" void kernel_launch(void* const* d_in, const int* in_sizes, int n_in,
                              void* d_out, int out_size, void* d_ws, size_t ws_size,
                              hipStream_t stream) {
  (void)in_sizes; (void)n_in; (void)out_size; (void)ws_size;
  const float* x = (const float*)d_in[0];
  float* out  = (float*)d_out;
  float* mm   = (float*)d_ws;                    // [128][32][2] = 8192 floats
  float* sums = mm + B_IMG * SLICES * 2;         // [128][32][3] = 12288 floats

  dim3 grid(B_IMG * SLICES);
  k_minmax<<<grid, THREADS, 0, stream>>>(x, mm);
  k_sums  <<<grid, THREADS, 0, stream>>>(x, mm, sums);
  k_final <<<1, 128, 0, stream>>>(sums, out);
}